// SqueezeLayer_2542620639873
// MI455X (gfx1250) — compile-verified
//
#include <hip/hip_runtime.h>

// Checkerboard space-to-depth (SqueezeLayer fwd):
//   out[b, 4c + 2*kh + kw, h2, w2] = x[b, c, 2*h2 + kh, 2*w2 + kw]
// Shapes fixed by the reference: x = (8, 64, 512, 512) fp32.
//
// Pure permutation, 0 FLOPs, 1 GiB total traffic -> HBM-bound (~46 us at
// 23.3 TB/s). Strategy: each thread owns 8 consecutive input floats
// (2x b128 loads, zero load-side index math since input is linear in tid),
// splits them by w-parity in registers, and writes two b128 stores to the
// two destination channel planes. Both loads and stores are fully
// wave-coalesced. Non-temporal hints on both sides: streams are 512 MiB
// each, far larger than the 192 MB L2, and never re-read.

typedef float v4f __attribute__((ext_vector_type(4)));

__global__ __launch_bounds__(256) void squeeze_s2d_kernel(
    const float* __restrict__ x, float* __restrict__ out) {
  // One thread per 8 consecutive input floats.
  const unsigned tid = blockIdx.x * 256u + threadIdx.x;   // [0, 2^24)

  // ---- Load side: input is exactly linear in tid (floats [8t, 8t+8)). ----
  const v4f* __restrict__ src = reinterpret_cast<const v4f*>(x) + (size_t)tid * 2u;
  v4f a = __builtin_nontemporal_load(src);       // w+0..w+3
  v4f b = __builtin_nontemporal_load(src + 1);   // w+4..w+7

  // ---- Decompose tid -> (b, c, h, wchunk); all dims are powers of two. ----
  const unsigned wchunk = tid & 63u;        // which 8-float chunk in the W=512 row
  const unsigned rest   = tid >> 6;
  const unsigned h      = rest & 511u;      // H = 512
  const unsigned bc     = rest >> 9;        // b*64 + c  (C = 64)
  const unsigned kh     = h & 1u;
  const unsigned h2     = h >> 1;
  const unsigned bidx   = bc >> 6;
  const unsigned c      = bc & 63u;

  // Global output channel index: b*256 + 4c + 2*kh  (kw=0 plane; kw=1 is +1).
  const unsigned oc0 = (bidx << 8) | (c << 2) | (kh << 1);

  // out index (floats): ((oc0 * 256 + h2) * 256 + wchunk*4); H2 = W2 = 256.
  const size_t out0 = ((size_t)oc0 * 256u + h2) * 256u + (size_t)wchunk * 4u;

  // ---- Parity split in registers (the whole "conv"). ----
  v4f even = {a.x, a.z, b.x, b.z};   // kw = 0
  v4f odd  = {a.y, a.w, b.y, b.w};   // kw = 1

  // ---- Store side: adjacent lanes -> adjacent 16B chunks, 512B/wave/store.
  __builtin_nontemporal_store(even, reinterpret_cast<v4f*>(out + out0));
  __builtin_nontemporal_store(odd,
      reinterpret_cast<v4f*>(out + out0 + 256u * 256u));  // next channel plane
}

extern "C" void kernel_launch(void* const* d_in, const int* in_sizes, int n_in,
                              void* d_out, int out_size, void* d_ws, size_t ws_size,
                              hipStream_t stream) {
  const float* x = (const float*)d_in[0];
  float* out = (float*)d_out;

  const int n = in_sizes[0];            // 8*64*512*512 = 134217728 floats
  const int threads_total = n / 8;      // 8 floats per thread
  const int block = 256;                // 8 wave32 waves per block
  const int grid = threads_total / block;

  squeeze_s2d_kernel<<<grid, block, 0, stream>>>(x, out);
}